// GraphSAGENet_69097433858679
// MI455X (gfx1250) — compile-verified
//
#include <hip/hip_runtime.h>

typedef __attribute__((ext_vector_type(2))) float v2f;
typedef __attribute__((ext_vector_type(8))) float v8f;

#define NFEAT 128
#define NHID  16
#define NCLS  40

// ---------------- utility kernels ----------------

__global__ void zero_f32(float* __restrict__ p, long n) {
  long i = (long)blockIdx.x * blockDim.x + threadIdx.x;
  long s = (long)gridDim.x * blockDim.x;
  for (; i < n; i += s) p[i] = 0.0f;
}

__global__ void degree_kernel(const int* __restrict__ dst, int E, float* __restrict__ deg) {
  int e = blockIdx.x * blockDim.x + threadIdx.x;
  if (e < E) atomicAdd(&deg[dst[e]], 1.0f);
}

// scatter-add 16-wide rows: agg[dst[e]] += feat[src[e]]  (thread per edge*feature)
__global__ void edge_agg16(const int* __restrict__ src, const int* __restrict__ dst,
                           const float* __restrict__ feat, float* __restrict__ agg, int E) {
  long t = (long)blockIdx.x * blockDim.x + threadIdx.x;
  long total = (long)E * NHID;
  if (t >= total) return;
  int e = (int)(t >> 4);
  int f = (int)(t & 15);
  int s = src[e];
  int d = dst[e];
  atomicAdd(&agg[(long)d * NHID + f], feat[(long)s * NHID + f]);
}

// elementwise layer-1 epilogue: h = relu(s1 + agg1/max(deg,1) + b1)
__global__ void layer1_epilogue(const float* __restrict__ s1, const float* __restrict__ agg1,
                                const float* __restrict__ deg, const float* __restrict__ b1,
                                float* __restrict__ h, int N) {
  long t = (long)blockIdx.x * blockDim.x + threadIdx.x;
  long total = (long)N * NHID;
  if (t >= total) return;
  int node = (int)(t >> 4);
  int f = (int)(t & 15);
  float inv = 1.0f / fmaxf(deg[node], 1.0f);
  float v = s1[t] + agg1[t] * inv + b1[f];
  h[t] = fmaxf(v, 0.0f);
}

// ---------------- WMMA GEMM kernels (fp32, 16x16x4) ----------------
// One wave handles a 16-row node tile. A: 16x4 fp32 (2 VGPRs/lane),
// lanes 0-15 carry K={k,k+1} of row M=lane, lanes 16-31 carry K={k+2,k+3}.
// B: 4x16, lane%16 = output column. C/D: v8f, VGPR r -> M = r + 8*(lane>=16).

// Dual-output projection: y = x @ wn, s = x @ wsf  (shared A-matrix loads).
__global__ void gemm128_dual_wmma(const float* __restrict__ x, const float* __restrict__ wn,
                                  const float* __restrict__ wsf,
                                  float* __restrict__ y, float* __restrict__ s, int N) {
  int lane = threadIdx.x & 31;
  int wave = threadIdx.x >> 5;
  int tile = blockIdx.x * (blockDim.x >> 5) + wave;
  int base = tile * 16;
  if (base >= N) return;                 // wave-uniform
  int mcol = lane & 15;
  int hi   = lane >> 4;
  int kb   = hi * 2;
  int row  = base + mcol;
  int rowc = row < N ? row : N - 1;      // clamp loads, guard stores
  const float* xr = x + (long)rowc * NFEAT;

  v8f accN = {};
  v8f accS = {};
#pragma unroll 8
  for (int k0 = 0; k0 < NFEAT; k0 += 4) {
    int k = k0 + kb;
    v2f a, bn, bs;
    a.x  = xr[k];
    a.y  = xr[k + 1];
    bn.x = wn[k * NHID + mcol];
    bn.y = wn[(k + 1) * NHID + mcol];
    bs.x = wsf[k * NHID + mcol];
    bs.y = wsf[(k + 1) * NHID + mcol];
    accN = __builtin_amdgcn_wmma_f32_16x16x4_f32(false, a, false, bn,
                                                 (short)0, accN, false, false);
    accS = __builtin_amdgcn_wmma_f32_16x16x4_f32(false, a, false, bs,
                                                 (short)0, accS, false, false);
  }

  if (base + 16 <= N) {                  // wave-uniform fast path: no per-lane guards
#pragma unroll
    for (int r = 0; r < 8; ++r) {
      long m = base + r + hi * 8;
      y[m * NHID + mcol] = accN[r];
      s[m * NHID + mcol] = accS[r];
    }
  } else {
#pragma unroll
    for (int r = 0; r < 8; ++r) {
      int m = base + r + hi * 8;
      if (m < N) {
        y[(long)m * NHID + mcol] = accN[r];
        s[(long)m * NHID + mcol] = accS[r];
      }
    }
  }
}

// out[N,40] = h @ w_self2 + (agg2/max(deg,1)) @ w_neigh2 + b2
// Treated as [h | mean] (16x32) @ [Ws2; Wn2] (32x40), K=32, 3 column tiles.
__global__ void layer2_out_wmma(const float* __restrict__ h, const float* __restrict__ agg2,
                                const float* __restrict__ deg,
                                const float* __restrict__ ws2, const float* __restrict__ wn2,
                                const float* __restrict__ b2, float* __restrict__ out, int N) {
  int lane = threadIdx.x & 31;
  int wave = threadIdx.x >> 5;
  int tile = blockIdx.x * (blockDim.x >> 5) + wave;
  int base = tile * 16;
  if (base >= N) return;
  int mcol = lane & 15;
  int hi   = lane >> 4;
  int kb   = hi * 2;
  int row  = base + mcol;
  int rowc = row < N ? row : N - 1;
  const float* hr = h    + (long)rowc * NHID;
  const float* ar = agg2 + (long)rowc * NHID;
  float inv = 1.0f / fmaxf(deg[rowc], 1.0f);
  bool fullTile = (base + 16 <= N);      // wave-uniform

  for (int ct = 0; ct < 3; ++ct) {       // wave-uniform loop
    int colBase = ct * 16;
    int col  = colBase + mcol;
    int colc = col < NCLS ? col : NCLS - 1;
    v8f acc = {};
#pragma unroll
    for (int k0 = 0; k0 < 2 * NHID; k0 += 4) {
      int k = k0 + kb;
      v2f a, b;
      if (k0 < NHID) {                   // compile-time uniform after unroll
        a.x = hr[k];
        a.y = hr[k + 1];
        b.x = ws2[k * NCLS + colc];
        b.y = ws2[(k + 1) * NCLS + colc];
      } else {
        int kk = k - NHID;
        a.x = ar[kk] * inv;
        a.y = ar[kk + 1] * inv;
        b.x = wn2[kk * NCLS + colc];
        b.y = wn2[(kk + 1) * NCLS + colc];
      }
      acc = __builtin_amdgcn_wmma_f32_16x16x4_f32(false, a, false, b,
                                                  (short)0, acc, false, false);
    }
    float bias = b2[colc];
    bool colOk = (col < NCLS);           // only col-tile 2 has ragged columns
    if (fullTile && ct < 2) {            // wave-uniform fast path
#pragma unroll
      for (int r = 0; r < 8; ++r) {
        long m = base + r + hi * 8;
        out[m * NCLS + col] = acc[r] + bias;
      }
    } else {
#pragma unroll
      for (int r = 0; r < 8; ++r) {
        int m = base + r + hi * 8;
        if (m < N && colOk) out[(long)m * NCLS + col] = acc[r] + bias;
      }
    }
  }
}

// ---------------- launch ----------------

extern "C" void kernel_launch(void* const* d_in, const int* in_sizes, int n_in,
                              void* d_out, int out_size, void* d_ws, size_t ws_size,
                              hipStream_t stream) {
  const float* x   = (const float*)d_in[0];
  const int*   src = (const int*)d_in[1];
  const int*   dst = (const int*)d_in[2];
  const float* wn1 = (const float*)d_in[3];
  const float* ws1 = (const float*)d_in[4];
  const float* b1  = (const float*)d_in[5];
  const float* wn2 = (const float*)d_in[6];
  const float* ws2 = (const float*)d_in[7];
  const float* b2  = (const float*)d_in[8];
  float* out = (float*)d_out;

  int N = in_sizes[0] / NFEAT;
  int E = in_sizes[1];

  float* ws   = (float*)d_ws;
  float* deg  = ws;                            // [N]
  float* agg1 = deg  + (long)N;                // [N,16]
  float* agg2 = agg1 + (long)N * NHID;         // [N,16]
  float* y1   = agg2 + (long)N * NHID;         // [N,16]
  float* s1   = y1   + (long)N * NHID;         // [N,16]
  float* hbuf = s1   + (long)N * NHID;         // [N,16]

  // zero deg + agg1 + agg2 (contiguous)
  long nzero = (long)N * (1 + 2 * NHID);
  zero_f32<<<512, 256, 0, stream>>>(ws, nzero);

  degree_kernel<<<(E + 255) / 256, 256, 0, stream>>>(dst, E, deg);

  int gemmBlocks = (N + 127) / 128;            // 8 waves/block, 16 rows/wave

  // layer 1: project both branches in one x pass (segment_sum is linear)
  gemm128_dual_wmma<<<gemmBlocks, 256, 0, stream>>>(x, wn1, ws1, y1, s1, N);

  long eaT = (long)E * NHID;
  int  eaB = (int)((eaT + 255) / 256);
  edge_agg16<<<eaB, 256, 0, stream>>>(src, dst, y1, agg1, E);

  long neT = (long)N * NHID;
  int  neB = (int)((neT + 255) / 256);
  layer1_epilogue<<<neB, 256, 0, stream>>>(s1, agg1, deg, b1, hbuf, N);

  // layer 2: aggregate h first (16-wide), project after
  edge_agg16<<<eaB, 256, 0, stream>>>(src, dst, hbuf, agg2, E);

  layer2_out_wmma<<<gemmBlocks, 256, 0, stream>>>(hbuf, agg2, deg, ws2, wn2, b2, out, N);
}